// GAT_Model_18167711662672
// MI455X (gfx1250) — compile-verified
//
#include <hip/hip_runtime.h>
#include <hip/hip_bf16.h>
#include <math.h>

#define HDIM 128
#define NEG_SLOPE 0.2f

typedef __attribute__((ext_vector_type(2))) float v2f;
typedef __attribute__((ext_vector_type(8))) float v8f;

// ---------------------------------------------------------------------------
// fp32 WMMA GEMM: Out[nrows x ncols] = X[nrows x K] @ W[K x ncols] (+ bias)
// One wave -> one 16x16 output tile via V_WMMA_F32_16X16X4_F32, K-loop step 4.
// Block = 256 threads = 8 waves = 16 rows x 128 cols of output.
// ---------------------------------------------------------------------------
__global__ __launch_bounds__(256) void wmma_gemm_f32(
    const float* __restrict__ X, const float* __restrict__ W,
    const float* __restrict__ bias, float* __restrict__ Out,
    int nrows, int ncols, int K, int has_bias)
{
  const int lane = threadIdx.x & 31;
  const int wave = threadIdx.x >> 5;
  const int row0 = blockIdx.x * 16;
  const int col0 = (blockIdx.y * 8 + wave) * 16;
  if (row0 >= nrows || col0 >= ncols) return;   // wave-uniform: EXEC stays all-1s

  // V_WMMA_F32_16X16X4_F32 operand layout (wave32, ISA 7.12.2):
  //  A (16x4):   lanes 0-15: M=lane,    {v0,v1}={K0,K1}; lanes 16-31: M=lane-16, {K2,K3}
  //  B (4x16):   lanes 0-15: N=lane,    {v0,v1}={K0,K1}; lanes 16-31: N=lane-16, {K2,K3}
  //  C/D(16x16): VGPR v: lanes 0-15 -> M=v ; lanes 16-31 -> M=v+8 ; N = lane&15
  const int mn  = lane & 15;
  const int khi = (lane >> 4) << 1;             // 0 or 2
  int arow = row0 + mn; if (arow >= nrows) arow = nrows - 1;   // clamp (discarded later)
  int bcol = col0 + mn; if (bcol >= ncols) bcol = ncols - 1;
  const float* Ap = X + (size_t)arow * K;

  v8f c = {};
  for (int k0 = 0; k0 < K; k0 += 4) {
    v2f a, b;
    a.x = Ap[k0 + khi];
    a.y = Ap[k0 + khi + 1];
    b.x = W[(size_t)(k0 + khi) * ncols + bcol];
    b.y = W[(size_t)(k0 + khi + 1) * ncols + bcol];
    c = __builtin_amdgcn_wmma_f32_16x16x4_f32(false, a, false, b, (short)0, c,
                                              false, false);
  }

  const int outCol = col0 + mn;
  if (outCol >= ncols) return;
  const float badd = has_bias ? bias[outCol] : 0.0f;
  const int rbase = row0 + ((lane >> 4) << 3);  // +0 (lanes 0-15) or +8 (lanes 16-31)
#pragma unroll
  for (int v = 0; v < 8; ++v) {
    const int r = rbase + v;
    if (r < nrows) Out[(size_t)r * ncols + outCol] = c[v] + badd;
  }
}

// ---------------------------------------------------------------------------
// es[i] = h[i,:].a_src ; ed[i] = h[i,:].a_dst   (one wave32 per node)
// ---------------------------------------------------------------------------
__global__ __launch_bounds__(256) void attn_scores(
    const float* __restrict__ h, const float* __restrict__ a_s,
    const float* __restrict__ a_d, float* __restrict__ es,
    float* __restrict__ ed, int n)
{
  const int node = (int)((blockIdx.x * (size_t)blockDim.x + threadIdx.x) >> 5);
  const int lane = threadIdx.x & 31;
  if (node >= n) return;
  const float* hp = h + (size_t)node * HDIM;
  float ss = 0.f, sd = 0.f;
#pragma unroll
  for (int j = 0; j < HDIM; j += 32) {
    const float v = hp[j + lane];
    ss += v * a_s[j + lane];
    sd += v * a_d[j + lane];
  }
#pragma unroll
  for (int o = 16; o > 0; o >>= 1) {
    ss += __shfl_xor(ss, o, 32);
    sd += __shfl_xor(sd, o, 32);
  }
  if (lane == 0) { es[node] = ss; ed[node] = sd; }
}

// ---------------------------------------------------------------------------
// Per-layer segment state init: agg=0, m=-inf, den=0
// ---------------------------------------------------------------------------
__global__ __launch_bounds__(256) void init_seg(
    float* __restrict__ agg, float* __restrict__ m, float* __restrict__ den,
    int n, size_t nh)
{
  const size_t i = blockIdx.x * (size_t)blockDim.x + threadIdx.x;
  if (i < nh) agg[i] = 0.0f;
  if (i < (size_t)n) { m[i] = -__builtin_inff(); den[i] = 0.0f; }
}

// float atomic-max via ordered-int trick (valid with -inf initialization)
__device__ __forceinline__ void atomicMaxFloat(float* addr, float val) {
  if (val >= 0.0f) atomicMax((int*)addr, __float_as_int(val));
  else             atomicMin((unsigned int*)addr, __float_as_uint(val));
}

// ---------------------------------------------------------------------------
// Pass 1: e = leaky_relu(es[src]+ed[dst]); ew[e]=e; m[dst]=max(m[dst],e)
// Edges e >= E are the implicit self loops (src=dst=e-E).
// ---------------------------------------------------------------------------
__global__ __launch_bounds__(256) void edge_score_max(
    const int* __restrict__ src, const int* __restrict__ dst, int E, int EE,
    const float* __restrict__ es, const float* __restrict__ ed,
    float* __restrict__ ew, float* __restrict__ m)
{
  const int e = (int)(blockIdx.x * (size_t)blockDim.x + threadIdx.x);
  if (e >= EE) return;
  int s, d;
  if (e < E) { s = src[e]; d = dst[e]; } else { s = d = e - E; }
  float v = es[s] + ed[d];
  v = (v > 0.0f) ? v : v * NEG_SLOPE;
  ew[e] = v;
  atomicMaxFloat(&m[d], v);
}

// ---------------------------------------------------------------------------
// Pass 2: w = exp(e - m[dst]); ew[e]=w; den[dst]+=w
// ---------------------------------------------------------------------------
__global__ __launch_bounds__(256) void edge_exp_sum(
    const int* __restrict__ dst, int E, int EE,
    float* __restrict__ ew, const float* __restrict__ m, float* __restrict__ den)
{
  const int e = (int)(blockIdx.x * (size_t)blockDim.x + threadIdx.x);
  if (e >= EE) return;
  const int d = (e < E) ? dst[e] : (e - E);
  const float w = expf(ew[e] - m[d]);
  ew[e] = w;
  atomicAdd(&den[d], w);
}

// ---------------------------------------------------------------------------
// Pass 3: agg[dst,:] += (w/den[dst]) * h[src,:]   (one wave32 per edge,
// each lane handles 4 coalesced floats -> 128B bursts per step)
// ---------------------------------------------------------------------------
__global__ __launch_bounds__(256) void edge_aggregate(
    const int* __restrict__ src, const int* __restrict__ dst, int E, int EE,
    const float* __restrict__ ew, const float* __restrict__ den,
    const float* __restrict__ h, float* __restrict__ agg)
{
  const int e = (int)((blockIdx.x * (size_t)blockDim.x + threadIdx.x) >> 5);
  const int lane = threadIdx.x & 31;
  if (e >= EE) return;
  int s, d;
  if (e < E) { s = src[e]; d = dst[e]; } else { s = d = e - E; }
  const float alpha = ew[e] / den[d];
  const float* hs = h + (size_t)s * HDIM;
  float* ad = agg + (size_t)d * HDIM;
#pragma unroll
  for (int j = 0; j < HDIM; j += 32)
    atomicAdd(&ad[j + lane], alpha * hs[j + lane]);
}

// ---------------------------------------------------------------------------
// x_next = relu(agg + b)  (in place)
// ---------------------------------------------------------------------------
__global__ __launch_bounds__(256) void bias_relu(
    float* __restrict__ xio, const float* __restrict__ b, size_t nh)
{
  const size_t i = blockIdx.x * (size_t)blockDim.x + threadIdx.x;
  if (i >= nh) return;
  const float v = xio[i] + b[i & (HDIM - 1)];
  xio[i] = (v > 0.0f) ? v : 0.0f;
}

// ---------------------------------------------------------------------------
extern "C" void kernel_launch(void* const* d_in, const int* in_sizes, int n_in,
                              void* d_out, int out_size, void* d_ws, size_t ws_size,
                              hipStream_t stream)
{
  (void)n_in; (void)out_size; (void)ws_size;

  const float* x  = (const float*)d_in[0];
  const int*   ei = (const int*)d_in[1];
  const int N  = in_sizes[0] / HDIM;    // 50000
  const int E  = in_sizes[1] / 2;       // 1,600,000
  const int EE = E + N;                 // + self loops
  const int* src = ei;                  // edge_index[0,:]
  const int* dst = ei + E;              // edge_index[1,:]

  const float* Wl[3]  = {(const float*)d_in[2], (const float*)d_in[6],  (const float*)d_in[10]};
  const float* asl[3] = {(const float*)d_in[3], (const float*)d_in[7],  (const float*)d_in[11]};
  const float* adl[3] = {(const float*)d_in[4], (const float*)d_in[8],  (const float*)d_in[12]};
  const float* bl[3]  = {(const float*)d_in[5], (const float*)d_in[9],  (const float*)d_in[13]};
  const float* Wo = (const float*)d_in[14];
  const float* bo = (const float*)d_in[15];
  const int C = in_sizes[15];           // 40

  // Workspace layout (fp32): two N*H ping-pong buffers + segment state + edge weights
  float* ws = (float*)d_ws;
  const size_t NH = (size_t)N * HDIM;
  float* h   = ws;                      // post-GEMM features
  float* agg = ws + NH;                 // aggregation / next-layer input
  float* es  = ws + 2 * NH;
  float* ed  = es + N;
  float* m   = ed + N;
  float* den = m + N;
  float* ew  = den + N;                 // EE floats

  const unsigned gRowTiles = (unsigned)((N + 15) / 16);
  const unsigned gNode  = (unsigned)(((size_t)N * 32 + 255) / 256);
  const unsigned gNH    = (unsigned)((NH + 255) / 256);
  const unsigned gEdge  = (unsigned)((EE + 255) / 256);
  const unsigned gEdgeW = (unsigned)(((size_t)EE * 32 + 255) / 256);

  const float* xin = x;
  for (int l = 0; l < 3; ++l) {
    wmma_gemm_f32<<<dim3(gRowTiles, 1), 256, 0, stream>>>(xin, Wl[l], nullptr, h,
                                                          N, HDIM, HDIM, 0);
    attn_scores<<<gNode, 256, 0, stream>>>(h, asl[l], adl[l], es, ed, N);
    init_seg<<<gNH, 256, 0, stream>>>(agg, m, den, N, NH);
    edge_score_max<<<gEdge, 256, 0, stream>>>(src, dst, E, EE, es, ed, ew, m);
    edge_exp_sum<<<gEdge, 256, 0, stream>>>(dst, E, EE, ew, m, den);
    edge_aggregate<<<gEdgeW, 256, 0, stream>>>(src, dst, E, EE, ew, den, h, agg);
    bias_relu<<<gNH, 256, 0, stream>>>(agg, bl[l], NH);
    xin = agg;
  }

  // Final projection: out = h3 @ Wo + bo  -> [N, 40]
  wmma_gemm_f32<<<dim3(gRowTiles, 1), 256, 0, stream>>>(xin, Wo, bo, (float*)d_out,
                                                        N, C, HDIM, 1);
}